// Attention_Rotary_Embedding_XL_46523085750473
// MI455X (gfx1250) — compile-verified
//
#include <hip/hip_runtime.h>
#include <math.h>

// ---------------------------------------------------------------------------
// Attention + RoPE + XL-memory block for MI455X (gfx1250, wave32, WMMA).
// fp32 math via V_WMMA_F32_16X16X4_F32 (matches fp32 reference; problem is
// HBM/L2 bound at 23.3 TB/s so the fp32 matrix path costs nothing).
// GEMMs use LDS-staged tiles fed by GLOBAL_LOAD_ASYNC_TO_LDS_B128 (ASYNCcnt)
// with double buffering so global traffic overlaps WMMA compute.
// ---------------------------------------------------------------------------

typedef __attribute__((ext_vector_type(2))) float v2f;
typedef __attribute__((ext_vector_type(4))) float v4f;
typedef __attribute__((ext_vector_type(8))) float v8f;
typedef int v4i_vec __attribute__((vector_size(16)));   // builtin's element type

#define D_MODEL  2048
#define N_HEADS  16
#define HEAD_DIM 128
#define BATCH    2
#define SEQ      1024
#define XL_LEN   1024
#define KV_LEN   (XL_LEN + SEQ)   // 2048

__device__ __forceinline__ v8f wmma4(v2f a, v2f b, v8f c) {
  // 8 args: (neg_a, A, neg_b, B, c_mod, C, reuse_a, reuse_b)
  return __builtin_amdgcn_wmma_f32_16x16x4_f32(false, a, false, b, (short)0, c,
                                               false, false);
}

__device__ __forceinline__ v8f zero8() {
  v8f z = {0.f, 0.f, 0.f, 0.f, 0.f, 0.f, 0.f, 0.f};
  return z;
}

// Async 16-byte global->LDS copy (gfx1250 GLOBAL_LOAD_ASYNC_TO_LDS_B128,
// tracked by ASYNCcnt). Builtin signature: (v4i as1* src, v4i as3* dst,
// imm offset, imm cpol). Falls back to a synchronous copy if unavailable.
__device__ __forceinline__ void async_copy16(float* lds_dst, const float* gsrc) {
#if __has_builtin(__builtin_amdgcn_global_load_async_to_lds_b128)
  __builtin_amdgcn_global_load_async_to_lds_b128(
      (__attribute__((address_space(1))) v4i_vec*)gsrc,
      (__attribute__((address_space(3))) v4i_vec*)lds_dst, 0, 0);
#else
  *(v4f*)lds_dst = *(const v4f*)gsrc;
#endif
}

__device__ __forceinline__ void wait_async0() {
#if __has_builtin(__builtin_amdgcn_s_wait_asynccnt)
  __builtin_amdgcn_s_wait_asynccnt(0);
#else
  asm volatile("s_wait_asynccnt 0x0" ::: "memory");
#endif
}

// ---------------------------------------------------------------------------
// Block-tiled TN GEMM: C[M,N] = A[M,K] * B[N,K]^T, row-major fp32.
// Block = 256 threads (8 waves) -> 128(M) x 64(N) C tile, K-blocked by 32,
// double-buffered LDS with padded stride 36 (bank-conflict-free, 16B-aligned).
// Wave (4m x 2n grid) computes a 32x32 sub-tile = 4 WMMA accumulators.
// Fragment layouts (V_WMMA_F32_16X16X4_F32):
//   A (16x4): lane l -> A[l%16][k + 2*(l/16) + j]
//   B (4x16): lane l -> Brow[l%16][k + 2*(l/16) + j]   (B stored N x K)
//   C:        lane l, vgpr rr -> row rr + 8*(l/16), col l%16
// ---------------------------------------------------------------------------
#define BM 128
#define BN 64
#define BK 32
#define LDP 36   // padded LDS row stride (floats); mult of 4 for b128 stores

__global__ __launch_bounds__(256) void gemm_tn_f32(
    const float* __restrict__ A, const float* __restrict__ B,
    float* __restrict__ C, int M, int N, int K) {
  __shared__ __align__(16) float As[2][BM * LDP];
  __shared__ __align__(16) float Bs[2][BN * LDP];

  int t = (int)threadIdx.x;
  int nTiles = N / BN;
  int bm = (int)blockIdx.x / nTiles;
  int bn = (int)blockIdx.x - bm * nTiles;
  int m0 = bm * BM, n0 = bn * BN;

  int wave = t >> 5, lane = t & 31;
  int wm = wave >> 1, wn = wave & 1;   // 4 x 2 wave grid
  int g = lane >> 4, r = lane & 15;

  const float* Ab = A + (size_t)m0 * K;
  const float* Bb = B + (size_t)n0 * K;

  // Issue one K-block of async global->LDS copies (24 KB total per block).
  auto load_block = [&](int buf, int k0) {
#pragma unroll
    for (int p = 0; p < 4; ++p) {              // A: 128 rows x 32 cols
      int c = t + p * 256;
      int row = c >> 3, col = (c & 7) * 4;
      async_copy16(&As[buf][row * LDP + col], Ab + (size_t)row * K + k0 + col);
    }
#pragma unroll
    for (int p = 0; p < 2; ++p) {              // B: 64 rows x 32 cols
      int c = t + p * 256;
      int row = c >> 3, col = (c & 7) * 4;
      async_copy16(&Bs[buf][row * LDP + col], Bb + (size_t)row * K + k0 + col);
    }
  };

  v8f acc[2][2];
#pragma unroll
  for (int im = 0; im < 2; ++im)
#pragma unroll
    for (int in = 0; in < 2; ++in) acc[im][in] = zero8();

  load_block(0, 0);
  wait_async0();
  __syncthreads();

  int nkb = K / BK;
  for (int kb = 0; kb < nkb; ++kb) {
    int cur = kb & 1;
    if (kb + 1 < nkb) load_block(cur ^ 1, (kb + 1) * BK);  // prefetch overlaps

    const float* Ac = &As[cur][0];
    const float* Bc = &Bs[cur][0];
#pragma unroll
    for (int kk = 0; kk < BK; kk += 4) {
      v2f af0 = *(const v2f*)(Ac + (wm * 32 + r) * LDP + kk + 2 * g);
      v2f af1 = *(const v2f*)(Ac + (wm * 32 + 16 + r) * LDP + kk + 2 * g);
      v2f bf0 = *(const v2f*)(Bc + (wn * 32 + r) * LDP + kk + 2 * g);
      v2f bf1 = *(const v2f*)(Bc + (wn * 32 + 16 + r) * LDP + kk + 2 * g);
      acc[0][0] = wmma4(af0, bf0, acc[0][0]);
      acc[0][1] = wmma4(af0, bf1, acc[0][1]);
      acc[1][0] = wmma4(af1, bf0, acc[1][0]);
      acc[1][1] = wmma4(af1, bf1, acc[1][1]);
    }
    wait_async0();       // this wave's prefetch transfers done
    __syncthreads();     // everyone done reading cur + writing nxt
  }

#pragma unroll
  for (int im = 0; im < 2; ++im)
#pragma unroll
    for (int in = 0; in < 2; ++in) {
      float* crow = C + (size_t)(m0 + wm * 32 + im * 16 + 8 * g) * N +
                    n0 + wn * 32 + in * 16 + r;
#pragma unroll
      for (int rr = 0; rr < 8; ++rr) crow[(size_t)rr * N] = acc[im][in][rr];
    }
}

// ---------------------------------------------------------------------------
// Prep: from qkv = x @ Wqkv^T build
//   q_ws   (B,H,T,hd)   : RoPE'd q
//   k_full (B,H,Kv,hd)  : rows [0,XL) = k_xl + pos_emb, rows [XL,Kv) = RoPE'd k
//   vt     (B,H,hd,Kv)  : transposed v_full (P*V WMMA B-frags contiguous)
// ---------------------------------------------------------------------------
__global__ __launch_bounds__(256) void prep_kernel(
    const float* __restrict__ qkv, const float* __restrict__ cosb,
    const float* __restrict__ sinb, const float* __restrict__ k_xl,
    const float* __restrict__ v_xl, const float* __restrict__ pos_emb,
    float* __restrict__ q_ws, float* __restrict__ k_full,
    float* __restrict__ vt) {
  size_t tid = (size_t)blockIdx.x * blockDim.x + threadIdx.x;
  int d2 = (int)(tid & 63);
  size_t rest = tid >> 6;
  int kv = (int)(rest & 2047);
  rest >>= 11;
  int h = (int)(rest & 15);
  int b = (int)(rest >> 4);
  if (b >= BATCH) return;

  int d  = d2 * 2;
  int bh = b * N_HEADS + h;

  float k0, k1, vv0, vv1;
  if (kv < XL_LEN) {
    size_t src = ((size_t)b * XL_LEN + kv) * D_MODEL + h * HEAD_DIM + d;
    size_t pe  = (size_t)kv * D_MODEL + h * HEAD_DIM + d;
    k0  = k_xl[src]     + pos_emb[pe];
    k1  = k_xl[src + 1] + pos_emb[pe + 1];
    vv0 = v_xl[src];
    vv1 = v_xl[src + 1];
  } else {
    int t = kv - XL_LEN;
    size_t qb = ((size_t)b * SEQ + t) * (3 * D_MODEL) + h * HEAD_DIM + d;
    float c = cosb[t * (HEAD_DIM / 2) + d2];
    float s = sinb[t * (HEAD_DIM / 2) + d2];
    float x1 = qkv[qb + D_MODEL], x2 = qkv[qb + D_MODEL + 1];
    k0 = x1 * c - x2 * s;
    k1 = x1 * s + x2 * c;
    vv0 = qkv[qb + 2 * D_MODEL];
    vv1 = qkv[qb + 2 * D_MODEL + 1];
    float q1 = qkv[qb], q2 = qkv[qb + 1];
    size_t qo = ((size_t)bh * SEQ + t) * HEAD_DIM + d;
    q_ws[qo]     = q1 * c - q2 * s;
    q_ws[qo + 1] = q1 * s + q2 * c;
  }

  size_t kf = ((size_t)bh * KV_LEN + kv) * HEAD_DIM + d;
  k_full[kf]     = k0;
  k_full[kf + 1] = k1;
  size_t vo = ((size_t)bh * HEAD_DIM + d) * KV_LEN + kv;
  vt[vo]          = vv0;
  vt[vo + KV_LEN] = vv1;
}

// ---------------------------------------------------------------------------
// Flash attention, one wave per (b,h, 16-row q tile). Online softmax over
// 128 key-chunks of 16. S = q*k^T via 32 WMMAs (k-frags batched 8 at a time,
// 2 accumulators to break the WMMA RAW chain), P -> A-layout via a private
// per-wave LDS slab (same-wave DS ops are in-order), Y += P*V via 32 WMMAs.
// ---------------------------------------------------------------------------
__global__ __launch_bounds__(128) void flash_attn_kernel(
    const float* __restrict__ q_ws, const float* __restrict__ k_full,
    const float* __restrict__ vt, float* __restrict__ y_ws,
    const int* __restrict__ causal_flag) {
  __shared__ float pshare[4 * 256];   // 1 KB per wave, 4 waves/block

  int lane = (int)(threadIdx.x & 31);
  int wInB = (int)(threadIdx.x >> 5);
  int wid  = (int)blockIdx.x * 4 + wInB;
  int qt = wid & ((SEQ / 16) - 1);    // 64 q tiles
  int bh = wid >> 6;                  // [0, B*H)
  int b = bh >> 4, h = bh & 15;
  int g = lane >> 4, r = lane & 15;
  int causal = causal_flag[0];
  const float scale = 0.088388347648318447f;  // 1/sqrt(128)

  // Preload the 16x128 q tile as 32 A-fragments.
  const float* qrow = q_ws + ((size_t)bh * SEQ + qt * 16 + r) * HEAD_DIM + 2 * g;
  v2f qf[32];
#pragma unroll
  for (int kk = 0; kk < 32; ++kk) qf[kk] = *(const v2f*)(qrow + kk * 4);

  v8f yacc[8];
#pragma unroll
  for (int t = 0; t < 8; ++t) yacc[t] = zero8();
  float mrow[8], lrow[8];
#pragma unroll
  for (int rr = 0; rr < 8; ++rr) { mrow[rr] = -3.0e38f; lrow[rr] = 0.f; }

  float* lds = pshare + wInB * 256;
  const float* kbase = k_full + (size_t)bh * KV_LEN * HEAD_DIM;
  const float* vtb   = vt     + (size_t)bh * HEAD_DIM * KV_LEN;
  int qrow0 = qt * 16;

  for (int chunk = 0; chunk < KV_LEN / 16; ++chunk) {
    // ---- scores tile: S = q * k^T (batched loads, dual accumulators) ----
    v8f s0 = zero8(), s1 = zero8();
    const float* krow = kbase + (size_t)(chunk * 16 + r) * HEAD_DIM + 2 * g;
#pragma unroll
    for (int kk0 = 0; kk0 < 32; kk0 += 8) {
      v2f bk[8];
#pragma unroll
      for (int j = 0; j < 8; ++j) bk[j] = *(const v2f*)(krow + (kk0 + j) * 4);
#pragma unroll
      for (int j = 0; j < 8; j += 2) {
        s0 = wmma4(qf[kk0 + j], bk[j], s0);
        s1 = wmma4(qf[kk0 + j + 1], bk[j + 1], s1);
      }
    }
    v8f s = s0 + s1;

    // ---- online softmax (row-wise across 16 lanes of each half-wave) ----
    int col = chunk * 16 + r;
#pragma unroll
    for (int rr = 0; rr < 8; ++rr) {
      float v = s[rr] * scale;
      int row = qrow0 + rr + 8 * g;
      if (causal && (col > row + (KV_LEN - SEQ))) v = -3.0e38f;
      float mx = v;
      mx = fmaxf(mx, __shfl_xor(mx, 1, 32));
      mx = fmaxf(mx, __shfl_xor(mx, 2, 32));
      mx = fmaxf(mx, __shfl_xor(mx, 4, 32));
      mx = fmaxf(mx, __shfl_xor(mx, 8, 32));
      float mnew = fmaxf(mrow[rr], mx);
      float corr = __expf(mrow[rr] - mnew);
      float p = __expf(v - mnew);
      float ps = p;
      ps += __shfl_xor(ps, 1, 32);
      ps += __shfl_xor(ps, 2, 32);
      ps += __shfl_xor(ps, 4, 32);
      ps += __shfl_xor(ps, 8, 32);
      lrow[rr] = lrow[rr] * corr + ps;
      mrow[rr] = mnew;
#pragma unroll
      for (int t = 0; t < 8; ++t) yacc[t][rr] *= corr;
      lds[(rr + 8 * g) * 16 + r] = p;  // C-layout -> LDS
    }

    // ---- P fragments in A-layout (same wave: DS ops are in-order) ----
    v2f pf[4];
#pragma unroll
    for (int c = 0; c < 4; ++c)
      pf[c] = *(const v2f*)(lds + r * 16 + 4 * c + 2 * g);

    // ---- Y += P * V  (batched vt loads; contiguous in kv) ----
#pragma unroll
    for (int tt = 0; tt < 8; tt += 2) {
      const float* vrow0 = vtb + (size_t)(tt * 16 + r) * KV_LEN + chunk * 16 + 2 * g;
      const float* vrow1 = vrow0 + (size_t)16 * KV_LEN;
      v2f bv[8];
#pragma unroll
      for (int c = 0; c < 4; ++c) {
        bv[c]     = *(const v2f*)(vrow0 + 4 * c);
        bv[4 + c] = *(const v2f*)(vrow1 + 4 * c);
      }
      v8f a0 = yacc[tt], a1 = yacc[tt + 1];
#pragma unroll
      for (int c = 0; c < 4; ++c) {
        a0 = wmma4(pf[c], bv[c], a0);
        a1 = wmma4(pf[c], bv[4 + c], a1);
      }
      yacc[tt] = a0;
      yacc[tt + 1] = a1;
    }
  }

  // ---- normalize and store in (B,T,C) layout for the output projection ----
#pragma unroll
  for (int rr = 0; rr < 8; ++rr) {
    float inv = 1.0f / lrow[rr];
    int row = qrow0 + rr + 8 * g;
    size_t base = ((size_t)b * SEQ + row) * D_MODEL + h * HEAD_DIM + r;
#pragma unroll
    for (int t = 0; t < 8; ++t) y_ws[base + t * 16] = yacc[t][rr] * inv;
  }
}

// ---------------------------------------------------------------------------
extern "C" void kernel_launch(void* const* d_in, const int* in_sizes, int n_in,
                              void* d_out, int out_size, void* d_ws,
                              size_t ws_size, hipStream_t stream) {
  const float* x       = (const float*)d_in[0];
  const float* cosb    = (const float*)d_in[1];
  const float* sinb    = (const float*)d_in[2];
  const float* k_xl    = (const float*)d_in[3];
  const float* v_xl    = (const float*)d_in[4];
  const float* pos_emb = (const float*)d_in[5];
  const float* w_qkv   = (const float*)d_in[6];
  const float* w_proj  = (const float*)d_in[7];
  const int*   is_causal = (const int*)d_in[8];
  float* out = (float*)d_out;

  // Workspace layout (128 MB total):
  //   [0,48MB)   qkv  (B,T,3C)  -- reused as y_ws after prep consumes it
  //   [48,64MB)  q_ws (B,H,T,hd)
  //   [64,96MB)  k_full (B,H,Kv,hd)
  //   [96,128MB) vt (B,H,hd,Kv)
  char* ws = (char*)d_ws;
  float* qkv    = (float*)ws;
  float* q_ws   = (float*)(ws + (size_t)48 * 1024 * 1024);
  float* k_full = (float*)(ws + (size_t)64 * 1024 * 1024);
  float* vt     = (float*)(ws + (size_t)96 * 1024 * 1024);
  float* y_ws   = qkv;  // safe reuse: attention no longer reads qkv

  // 1) QKV projection: M=2048, N=6144, K=2048
  {
    int M = BATCH * SEQ, N = 3 * D_MODEL, K = D_MODEL;
    int blocks = (M / BM) * (N / BN);
    gemm_tn_f32<<<dim3(blocks), dim3(256), 0, stream>>>(x, w_qkv, qkv, M, N, K);
  }
  // 2) RoPE + XL merge + v transpose
  {
    size_t total = (size_t)BATCH * N_HEADS * KV_LEN * (HEAD_DIM / 2);
    prep_kernel<<<dim3((unsigned)(total / 256)), dim3(256), 0, stream>>>(
        qkv, cosb, sinb, k_xl, v_xl, pos_emb, q_ws, k_full, vt);
  }
  // 3) Flash attention (one wave per 16-row q tile)
  {
    int waves = BATCH * N_HEADS * (SEQ / 16);  // 2048
    flash_attn_kernel<<<dim3(waves / 4), dim3(128), 0, stream>>>(
        q_ws, k_full, vt, y_ws, is_causal);
  }
  // 4) Output projection: M=2048, N=2048, K=2048
  {
    int M = BATCH * SEQ, N = D_MODEL, K = D_MODEL;
    int blocks = (M / BM) * (N / BN);
    gemm_tn_f32<<<dim3(blocks), dim3(256), 0, stream>>>(y_ws, w_proj, out, M, N, K);
  }
}